// HiddenElementsFinalLayer_81106162418203
// MI455X (gfx1250) — compile-verified
//
#include <hip/hip_runtime.h>

// out[i, h*D+d] = x[i,h] * W[h,d] + b[h,d]
// B=2048, H=1024, D=128. Pure streaming-store problem (1 GiB out @ 23.3 TB/s
// ~= 46 us floor). FMAs expressed as rank-1 V_WMMA_F32_16X16X4_F32 updates
// (K=1, bias passed as the C accumulator); output written with non-temporal
// stores so the 1 GiB stream doesn't evict the hot W/b/x working set from L2.
//
// Round-1 fix vs previous version: all lane masking is branchless (cndmask,
// no exec-mask save/restore), and the 16 W/b loads per wave are hoisted into
// one unrolled load block so the compiler can clause them and hide latency
// ahead of the WMMA + NT-store stream.

typedef __attribute__((ext_vector_type(2))) float v2f;
typedef __attribute__((ext_vector_type(8))) float v8f;

namespace {
constexpr int kB = 2048;
constexpr int kH = 1024;
constexpr int kD = 128;
}

__global__ __launch_bounds__(256) void bcast_fma_wmma_kernel(
    const float* __restrict__ x,     // [B, H]
    const float* __restrict__ W,     // [H, D]
    const float* __restrict__ bias,  // [H, D]
    float* __restrict__ out)         // [B, H*D]
{
    const int lane = threadIdx.x & 31;
    const int wave = threadIdx.x >> 5;
    const int nl   = lane & 15;
    const bool lo  = (lane < 16);

    const int h  = blockIdx.x * 8 + wave;  // one channel per wave
    const int i0 = blockIdx.y * 16;        // 16-row batch tile per block

    const float* Wrow = W    + (size_t)h * kD;
    const float* brow = bias + (size_t)h * kD;

    // ---- Hoisted loads: every lane reads the lane&15 element (lanes 16-31
    // duplicate the same 64B line -> still one coalesced request), masking to
    // zero happens later with cndmask, no exec divergence.
    float xv_all = x[(size_t)(i0 + nl) * kH + h];
    float wvals[8];
    float bvals[8];
#pragma unroll
    for (int dt = 0; dt < 8; ++dt) {
        wvals[dt] = Wrow[dt * 16 + nl];
        bvals[dt] = brow[dt * 16 + nl];
    }

    // ---- A matrix (16x4 f32): column K=0 holds x[i0..i0+15, h], K=1..3 = 0.
    // Lanes 0-15: VGPR0=K0 (x value), VGPR1=K1 (0); lanes 16-31: K2/K3 = 0.
    v2f A;
    A.x = lo ? xv_all : 0.0f;
    A.y = 0.0f;

    // Per-lane output base: row = i0 + 8*(lane>=16), col = h*D + (lane&15).
    // D-matrix VGPR r then lands at base + r*rowStride (uniform immediate).
    const size_t rowStride = (size_t)kH * kD;  // floats per output row
    float* obase = out + (size_t)(i0 + ((lane >> 4) << 3)) * rowStride
                       + (size_t)h * kD + nl;

#pragma unroll
    for (int dt = 0; dt < 8; ++dt) {
        const int d0 = dt * 16;

        // ---- B matrix (4x16 f32): VGPR0 lanes 0-15 = row K=0 = W[h,d0+N];
        // lanes 16-31 (K=2) and VGPR1 (K=1,3) zeroed via cndmask.
        v2f Bm;
        Bm.x = lo ? wvals[dt] : 0.0f;
        Bm.y = 0.0f;

        // ---- C matrix: C[M,N] = b[h, d0+N] for every M -> every one of the
        // 8 C VGPRs in a lane holds b[h, d0 + (lane&15)].
        const float bv = bvals[dt];
        v8f C = {bv, bv, bv, bv, bv, bv, bv, bv};

        // D = A x B + C  ==  x[i,h] * W[h,d] + b[h,d] for the 16x16 tile.
        v8f Dv = __builtin_amdgcn_wmma_f32_16x16x4_f32(
            /*neg_a=*/false, A, /*neg_b=*/false, Bm,
            /*c_mod=*/(short)0, C, /*reuse_a=*/false, /*reuse_b=*/false);

        // ---- Streaming non-temporal stores. VGPR r -> rows i0+r / i0+8+r,
        // 16 consecutive floats per half-wave (two 64B segments per instr).
        float* o = obase + d0;
#pragma unroll
        for (int r = 0; r < 8; ++r) {
            __builtin_nontemporal_store(Dv[r], o + (size_t)r * rowStride);
        }
    }
}

extern "C" void kernel_launch(void* const* d_in, const int* in_sizes, int n_in,
                              void* d_out, int out_size, void* d_ws, size_t ws_size,
                              hipStream_t stream) {
    (void)in_sizes; (void)n_in; (void)out_size; (void)d_ws; (void)ws_size;

    const float* x    = (const float*)d_in[0];  // [B, H]
    const float* W    = (const float*)d_in[1];  // [H, D]
    const float* bias = (const float*)d_in[2];  // [H, D]
    float* out        = (float*)d_out;          // [B, H*D]

    dim3 grid(kH / 8, kB / 16);  // (128, 128): 8 channels/block, 16 rows/block
    dim3 block(256);             // 8 waves of 32
    bcast_fma_wmma_kernel<<<grid, block, 0, stream>>>(x, W, bias, out);
}